// DynamicGraphConstruction_27204322853262
// MI455X (gfx1250) — compile-verified
//
#include <hip/hip_runtime.h>
#include <stdint.h>

#ifndef __has_builtin
#define __has_builtin(x) 0
#endif

// ---------------------------------------------------------------------------
// DynamicGraphConstruction for MI455X (gfx1250, wave32).
//   dist : [B,T,32,32] fp32  ->  out : [B,T,32,32] fp32
// One wave per 32x32 tile, one lane per row. Tile is staged global->LDS with
// CDNA5 async-to-LDS loads (ASYNCcnt), per-row top-4 via bitmask selection,
// degree/dinv shared through LDS within the wave.
// Roofline: 64MB in + 64MB out @ 23.3 TB/s ~ 5.5us; no matmul structure, so
// WMMA does not apply — async-to-LDS + wave32 lane mapping is the CDNA5 path.
// ---------------------------------------------------------------------------

static constexpr int kP         = 32;   // points per tile (== wave32 lanes)
static constexpr int kTileElems = kP * kP;
static constexpr int kTilesPerBlock = 8;    // 8 waves / 256 threads per block

typedef int v4i __attribute__((ext_vector_type(4)));
typedef __attribute__((address_space(1))) v4i  g_v4i;   // global v4i
typedef __attribute__((address_space(3))) v4i  l_v4i;   // LDS v4i
typedef __attribute__((address_space(1))) void gbl_void;
typedef __attribute__((address_space(3))) void lds_void;
typedef __attribute__((address_space(3))) char lds_char;

// Copy 16 bytes per lane from global to LDS asynchronously (ASYNCcnt++).
__device__ __forceinline__ void async_copy16(const void* gsrc, void* ldst) {
#if __has_builtin(__builtin_amdgcn_global_load_async_to_lds_b128)
  __builtin_amdgcn_global_load_async_to_lds_b128(
      (g_v4i*)(gbl_void*)(void*)gsrc, (l_v4i*)(lds_void*)ldst,
      /*offset=*/0, /*cpol=*/0);
#else
  unsigned loff = (unsigned)(uintptr_t)(lds_char*)ldst;   // 32-bit LDS offset
  uint64_t ga   = (uint64_t)(uintptr_t)gsrc;              // 64-bit global VA
  asm volatile("global_load_async_to_lds_b128 %0, %1, off"
               :
               : "v"(loff), "v"(ga)
               : "memory");
#endif
}

__device__ __forceinline__ void wait_async0() {
#if __has_builtin(__builtin_amdgcn_s_wait_asynccnt)
  __builtin_amdgcn_s_wait_asynccnt(0);
#else
  asm volatile("s_wait_asynccnt 0" ::: "memory");
#endif
  asm volatile("" ::: "memory");  // compiler fence: LDS reads stay below
}

__global__ __launch_bounds__(kTilesPerBlock * 32) void knn_norm_adj_kernel(
    const float* __restrict__ dist, float* __restrict__ out, int nTiles) {
  __shared__ alignas(16) float sTile[kTilesPerBlock][kTileElems];
  __shared__ alignas(16) float sDinv[kTilesPerBlock][kP];

  const int lane = threadIdx.x & 31;
  const int wave = threadIdx.x >> 5;
  const int tile = blockIdx.x * kTilesPerBlock + wave;
  if (tile >= nTiles) return;   // per-wave uniform; no block barriers used

  const size_t base = (size_t)tile * kTileElems;

  // ---- stage 4KB tile global -> LDS with 8 wave-wide async b128 loads ----
  const char* gbase = (const char*)(dist + base) + lane * 16;
  char*       lbase = (char*)&sTile[wave][0] + lane * 16;
#pragma unroll
  for (int j = 0; j < 8; ++j)
    async_copy16(gbase + j * 512, lbase + j * 512);

  wait_async0();  // ASYNCcnt -> 0 : tile resident in LDS for this wave

  // ---- lane p reads its 128B row from LDS ----
  float r[kP];
  {
    const float4* row = (const float4*)&sTile[wave][lane * kP];
#pragma unroll
    for (int j = 0; j < 8; ++j) {
      float4 v = row[j];
      r[4 * j + 0] = v.x; r[4 * j + 1] = v.y;
      r[4 * j + 2] = v.z; r[4 * j + 3] = v.w;
    }
  }

  // ---- top-4 smallest (diagonal excluded) via exclusion bitmask ----------
  // 'taken' doubles as the adjacency row: self-loop bit + 4 neighbor bits.
  unsigned taken = 1u << lane;
#pragma unroll
  for (int s = 0; s < 4; ++s) {
    float best = 3.0e38f;
    int   bi   = 0;
#pragma unroll
    for (int q = 0; q < kP; ++q) {
      const bool ok = (((taken >> q) & 1u) == 0u) && (r[q] < best);
      best = ok ? r[q] : best;
      bi   = ok ? q : bi;
    }
    taken |= 1u << bi;
  }

  // ---- degree / D^{-1/2} (degree is provably 5, computed generically) ----
  const float deg  = (float)__popc(taken);
  const float dinv = 1.0f / sqrtf(deg);
  sDinv[wave][lane] = dinv;
  __builtin_amdgcn_wave_barrier();  // keep the broadcast reads after the store
  // (same-wave LDS ops are executed in order by HW; this is a sched fence)

  // ---- emit normalized adjacency row: out[p][q] = dinv_p * dinv_q * A ----
  float* orow = out + base + (size_t)lane * kP;
  const float4* dv4 = (const float4*)&sDinv[wave][0];
#pragma unroll
  for (int j = 0; j < 8; ++j) {
    const float4 dv = dv4[j];
    float4 o;
    o.x = ((taken >> (4 * j + 0)) & 1u) ? dinv * dv.x : 0.0f;
    o.y = ((taken >> (4 * j + 1)) & 1u) ? dinv * dv.y : 0.0f;
    o.z = ((taken >> (4 * j + 2)) & 1u) ? dinv * dv.z : 0.0f;
    o.w = ((taken >> (4 * j + 3)) & 1u) ? dinv * dv.w : 0.0f;
    ((float4*)orow)[j] = o;  // global_store_b128, contiguous per lane
  }
}

extern "C" void kernel_launch(void* const* d_in, const int* in_sizes, int n_in,
                              void* d_out, int out_size, void* d_ws, size_t ws_size,
                              hipStream_t stream) {
  const float* dist = (const float*)d_in[0];
  float*       out  = (float*)d_out;
  const int nTiles  = in_sizes[0] / kTileElems;  // B*T = 16384
  const int blocks  = (nTiles + kTilesPerBlock - 1) / kTilesPerBlock;
  knn_norm_adj_kernel<<<blocks, kTilesPerBlock * 32, 0, stream>>>(dist, out, nTiles);
}